// GatedSpectralConv_27986006900971
// MI455X (gfx1250) — compile-verified
//
#include <hip/hip_runtime.h>
#include <stdint.h>

typedef __attribute__((ext_vector_type(2))) float v2f;
typedef __attribute__((ext_vector_type(8))) float v8f;
typedef __attribute__((ext_vector_type(4))) unsigned int u32x4;
typedef __attribute__((ext_vector_type(4))) int i32x4;
typedef __attribute__((ext_vector_type(8))) int i32x8;

#define TILE 64
#define KC   32
#define LDSS 36   // 32 + 4 pad DWORDs (TDM pad_interval=32dw, pad_amount=4dw): conflict-free, 16B aligned

// ---------------------------------------------------------------------------
// TDM: DMA one [64 rows x KC cols] f32 tile (row stride = K elements) into LDS
// with 4-DWORD padding after every 32 DWORDs -> LDS row stride LDSS.
// Descriptor layout per CDNA5 ISA 8.3/8.4 (2D tensor; groups 2/3 zero).
// ---------------------------------------------------------------------------
__device__ __forceinline__ void tdm_load_tile(uint32_t lds_addr, const float* gsrc, int K) {
    const uint64_t ga = (uint64_t)(uintptr_t)gsrc;
    u32x4 g0;
    g0[0] = 1u;                                   // count=1, user mode, no gather
    g0[1] = lds_addr;                             // LDS byte address
    g0[2] = (uint32_t)ga;                         // global_addr[31:0]
    g0[3] = (uint32_t)((ga >> 32) & 0x01FFFFFFull) | (2u << 30);  // addr[56:32], type=2
    i32x8 g1;
    g1[0] = (2 << 16)         // data_size = 4 bytes
          | (1 << 20)         // pad_enable
          | (4 << 22)         // pad_interval: 32 DWORDs
          | (3 << 25);        // pad_amount: 4 DWORDs
    g1[1] = (K & 0xFFFF) << 16;                        // tensor_dim0[15:0]
    g1[2] = ((unsigned)K >> 16) & 0xFFFF;              // tensor_dim0[31:16]; tensor_dim1 lo = 0
    g1[3] = 0x10 | (KC << 16);                         // tensor_dim1 = 1<<20 rows; tile_dim0 = KC
    g1[4] = TILE;                                      // tile_dim1 = 64 rows; tile_dim2 = 0
    g1[5] = K;                                         // tensor_dim0_stride[31:0]
    g1[6] = 0;
    g1[7] = 0;
    const i32x4 z4 = {0, 0, 0, 0};
#if defined(__clang_major__) && __clang_major__ >= 23
    const i32x8 z8 = {0, 0, 0, 0, 0, 0, 0, 0};
    __builtin_amdgcn_tensor_load_to_lds(g0, g1, z4, z4, z8, 0);
#else
    __builtin_amdgcn_tensor_load_to_lds(g0, g1, z4, z4, 0);
#endif
}

// ---------------------------------------------------------------------------
// Y[M,N] = X[M,K] * W[N,K]^T  (f32, row-major, M%64==0, N%64==0, K%32==0)
// 4 waves/block, 64x64 block tile, TDM double-buffered LDS staging,
// V_WMMA_F32_16X16X4_F32 compute.
// ---------------------------------------------------------------------------
__global__ __launch_bounds__(128)
void gemm_xt_wmma_f32(const float* __restrict__ X,
                      const float* __restrict__ W,
                      float* __restrict__ Y,
                      int K, int N) {
    __shared__ float xs[2][TILE * LDSS];
    __shared__ float wsh[2][TILE * LDSS];

    const int tid      = threadIdx.x;
    const int wave     = tid >> 5;
    const int lane     = tid & 31;
    const int lane_mod = lane & 15;
    const int lane_hi  = lane >> 4;
    const int m_blk    = blockIdx.y * TILE;
    const int n_blk    = blockIdx.x * TILE;

    v8f acc[4] = {};

    // Prologue: DMA tile 0 into buffer 0 (one issuing wave; TDM ignores EXEC).
    if (wave == 0) {
        tdm_load_tile((uint32_t)(uintptr_t)&xs[0][0],  X + (size_t)m_blk * K, K);
        tdm_load_tile((uint32_t)(uintptr_t)&wsh[0][0], W + (size_t)n_blk * K, K);
    }

    const int ntiles = K / KC;
    for (int it = 0; it < ntiles; ++it) {
        const int buf = it & 1;
        // Issuing wave waits for its DMA; others have TENSORcnt==0 (no-op).
        __builtin_amdgcn_s_wait_tensorcnt(0);
        // Publishes DMA'd data to all waves; also guarantees everyone finished
        // reading buf^1 (previous iteration) before we overwrite it below.
        __syncthreads();
        if (wave == 0 && it + 1 < ntiles) {
            const int kc = (it + 1) * KC;
            tdm_load_tile((uint32_t)(uintptr_t)&xs[buf ^ 1][0],  X + (size_t)m_blk * K + kc, K);
            tdm_load_tile((uint32_t)(uintptr_t)&wsh[buf ^ 1][0], W + (size_t)n_blk * K + kc, K);
        }
        const float* xb = xs[buf];
        const float* wb = wsh[buf];
        const int m_local = wave * 16 + lane_mod;
#pragma unroll
        for (int kk = 0; kk < KC; kk += 4) {
            // A frag: lanes 0-15 -> K = kk,kk+1 ; lanes 16-31 -> K = kk+2,kk+3
            const int kfrag = kk + (lane_hi << 1);
            v2f a;
            a.x = xb[m_local * LDSS + kfrag];
            a.y = xb[m_local * LDSS + kfrag + 1];
#pragma unroll
            for (int nt = 0; nt < 4; ++nt) {
                const int n_local = nt * 16 + lane_mod;
                v2f b;
                b.x = wb[n_local * LDSS + kfrag];
                b.y = wb[n_local * LDSS + kfrag + 1];
                acc[nt] = __builtin_amdgcn_wmma_f32_16x16x4_f32(
                    false, a, false, b, (short)0, acc[nt], false, false);
            }
        }
    }

    // C/D layout: VGPR r -> M = r + 8*lane_hi, N = lane_mod
    const int m_row0 = m_blk + wave * 16 + (lane_hi << 3);
#pragma unroll
    for (int nt = 0; nt < 4; ++nt) {
        const int n = n_blk + nt * 16 + lane_mod;
#pragma unroll
        for (int r = 0; r < 8; ++r) {
            Y[(size_t)(m_row0 + r) * N + n] = acc[nt][r];
        }
    }
}

// ---------------------------------------------------------------------------
// Chunked gated complex scan:  h_t = sigmoid(dec+bias)*rot * h_{t-1} + u_t
// Phase A: per-chunk (A_prod, U) summaries.  Phase B: sequential prefix over
// chunk summaries per (b,f) -> chunk initial states + last-state outputs.
// Phase C: replay each chunk from its initial state, writing y over spec.
// ---------------------------------------------------------------------------
__device__ __forceinline__ void gate_rot(float dlog, float bias, float rc, float rs,
                                         float& ar, float& ai) {
    const float g = 1.0f / (1.0f + __expf(-(dlog + bias)));
    ar = g * rc;
    ai = g * rs;
}

__global__ void scan_chunk_summary(const float* __restrict__ dec,
                                   const float* __restrict__ spec,
                                   const float* __restrict__ b_decay,
                                   const float* __restrict__ freqs,
                                   float4* __restrict__ csum,
                                   int B, int L, int F, int nch) {
    const int g = blockIdx.x * blockDim.x + threadIdx.x;
    if (g >= B * F * nch) return;
    const int f  = g % F;
    const int bc = g / F;
    const int c  = bc % nch;
    const int b  = bc / nch;
    const int tc = L / nch;

    const float bias  = b_decay[f];
    const float omega = freqs[f] * 0.1f;
    const float rc = cosf(omega);
    const float rs = sinf(omega);

    float Ar = 1.0f, Ai = 0.0f, Ur = 0.0f, Ui = 0.0f;
    const size_t row0 = (size_t)b * L + (size_t)c * tc;
    size_t decIdx = row0 * F + f;
    size_t spIdx  = row0 * (2 * F) + f;
    for (int t = 0; t < tc; ++t) {
        float ar, ai;
        gate_rot(dec[decIdx], bias, rc, rs, ar, ai);
        const float ur = spec[spIdx];
        const float ui = spec[spIdx + F];
        const float nUr = ar * Ur - ai * Ui + ur;
        const float nUi = ar * Ui + ai * Ur + ui;
        const float nAr = ar * Ar - ai * Ai;
        const float nAi = ar * Ai + ai * Ar;
        Ur = nUr; Ui = nUi; Ar = nAr; Ai = nAi;
        decIdx += F;
        spIdx  += 2 * F;
    }
    csum[g] = make_float4(Ar, Ai, Ur, Ui);
}

__global__ void scan_chunk_prefix(const float4* __restrict__ csum,
                                  float2* __restrict__ h0,
                                  float* __restrict__ last_out,
                                  int B, int F, int nch) {
    const int g = blockIdx.x * blockDim.x + threadIdx.x;
    if (g >= B * F) return;
    const int f = g % F;
    const int b = g / F;
    float hr = 0.0f, hi = 0.0f;
    for (int c = 0; c < nch; ++c) {
        const int idx = (b * nch + c) * F + f;
        h0[idx] = make_float2(hr, hi);
        const float4 s = csum[idx];   // (Ar, Ai, Ur, Ui)
        const float nr = s.x * hr - s.y * hi + s.z;
        const float ni = s.x * hi + s.y * hr + s.w;
        hr = nr; hi = ni;
    }
    last_out[g]         = hr;   // real(last) [B,F]
    last_out[B * F + g] = hi;   // imag(last) [B,F]
}

__global__ void scan_chunk_apply(const float* __restrict__ dec,
                                 float* __restrict__ spec,
                                 const float* __restrict__ b_decay,
                                 const float* __restrict__ freqs,
                                 const float2* __restrict__ h0,
                                 int B, int L, int F, int nch) {
    const int g = blockIdx.x * blockDim.x + threadIdx.x;
    if (g >= B * F * nch) return;
    const int f  = g % F;
    const int bc = g / F;
    const int c  = bc % nch;
    const int b  = bc / nch;
    const int tc = L / nch;

    const float bias  = b_decay[f];
    const float omega = freqs[f] * 0.1f;
    const float rc = cosf(omega);
    const float rs = sinf(omega);

    const float2 h = h0[g];
    float hr = h.x, hi = h.y;
    const size_t row0 = (size_t)b * L + (size_t)c * tc;
    size_t decIdx = row0 * F + f;
    size_t spIdx  = row0 * (2 * F) + f;
    for (int t = 0; t < tc; ++t) {
        float ar, ai;
        gate_rot(dec[decIdx], bias, rc, rs, ar, ai);
        const float ur = spec[spIdx];
        const float ui = spec[spIdx + F];
        const float nr = ar * hr - ai * hi + ur;
        const float ni = ar * hi + ai * hr + ui;
        hr = nr; hi = ni;
        spec[spIdx]     = hr;   // y_cat layout == u layout: [re | im]
        spec[spIdx + F] = hi;
        decIdx += F;
        spIdx  += 2 * F;
    }
}

// ---------------------------------------------------------------------------
extern "C" void kernel_launch(void* const* d_in, const int* in_sizes, int n_in,
                              void* d_out, int out_size, void* d_ws, size_t ws_size,
                              hipStream_t stream) {
    const float* x       = (const float*)d_in[0];  // [B,L,D]
    const float* W_spec  = (const float*)d_in[1];  // [2F,D]
    const float* W_decay = (const float*)d_in[2];  // [F,D]
    const float* b_decay = (const float*)d_in[3];  // [F]
    const float* freqs   = (const float*)d_in[4];  // [F]
    const float* W_out   = (const float*)d_in[5];  // [D,2F]
    float* out = (float*)d_out;

    const int B = 4, L = 4096, D = 1024, F = 512;
    const int M   = B * L;   // 16384 token rows
    const int NCH = 64;      // scan chunks (64 steps each)

    float*  dec  = (float*)d_ws;                        // [M, F]
    float*  spec = dec + (size_t)M * F;                 // [M, 2F]  (u, then y in place)
    float4* csum = (float4*)(spec + (size_t)M * 2 * F); // [B*F*NCH]
    float2* h0   = (float2*)(csum + (size_t)B * F * NCH);

    dim3 blk(128);
    // dec = x @ W_decay^T
    gemm_xt_wmma_f32<<<dim3(F / TILE, M / TILE), blk, 0, stream>>>(x, W_decay, dec, D, F);
    // spec = x @ W_spec^T
    gemm_xt_wmma_f32<<<dim3((2 * F) / TILE, M / TILE), blk, 0, stream>>>(x, W_spec, spec, D, 2 * F);
    // chunked gated scan
    const int nA = B * F * NCH;
    scan_chunk_summary<<<(nA + 255) / 256, 256, 0, stream>>>(dec, spec, b_decay, freqs,
                                                             csum, B, L, F, NCH);
    scan_chunk_prefix<<<(B * F + 255) / 256, 256, 0, stream>>>(csum, h0,
                                                               out + (size_t)M * D, B, F, NCH);
    scan_chunk_apply<<<(nA + 255) / 256, 256, 0, stream>>>(dec, spec, b_decay, freqs,
                                                           h0, B, L, F, NCH);
    // out = y_cat @ W_out^T
    gemm_xt_wmma_f32<<<dim3(D / TILE, M / TILE), blk, 0, stream>>>(spec, W_out, out, 2 * F, D);
}